// AssociativeMemoryBlock_5583457484857
// MI455X (gfx1250) — compile-verified
//
#include <hip/hip_runtime.h>
#include <hip/hip_bf16.h>
#include <math.h>

// ---------------------------------------------------------------------------
// AssociativeMemoryBlock for MI455X (gfx1250, wave32).
// P = prod_l (d_l I + c_l q_l q_l^T): chunked rank-1 recursion in LDS, then a
// binary tree of dense 128x128x128 fp32 WMMA GEMMs (V_WMMA_F32_16X16X4_F32),
// then Pq + epilogue. fp32 WMMA chosen for numerics (exp + 2049-long product);
// problem is latency-bound, not flop-bound, so fp32 matrix throughput is ample.
// ---------------------------------------------------------------------------

typedef __attribute__((ext_vector_type(2))) float v2f;
typedef __attribute__((ext_vector_type(8))) float v8f;

constexpr int DIM   = 128;
constexpr int SEQ   = 2048;
constexpr int LSEQ  = 2049;   // SEQ + 1 (prepended M_0 position)
constexpr int CHUNK = 64;
constexpr int SLOTS = 64;     // 33 real chunks padded to pow2 for the tree
constexpr int NB    = 2;

__device__ __forceinline__ v8f wmma4(v2f a, v2f b, v8f c) {
  // D(16x16,f32) += A(16x4,f32) * B(4x16,f32) + C
  return __builtin_amdgcn_wmma_f32_16x16x4_f32(
      false, a, false, b, (short)0, c, false, false);
}

// ---------------------------------------------------------------------------
// K1: projections q_exp/k_exp/v via WMMA, plus per-position scalars.
// grid = (NB*SEQ)/16 blocks of 256 threads (8 waves). Wave w -> N-tile w.
// ---------------------------------------------------------------------------
__global__ __launch_bounds__(256)
void k_proj(const float* __restrict__ x, const float* __restrict__ alpha_w,
            const float* __restrict__ Qw, const float* __restrict__ Kw,
            const float* __restrict__ Vw, const float* __restrict__ w0p,
            float* __restrict__ qe, float* __restrict__ ke, float* __restrict__ vv,
            float* __restrict__ wa, float* __restrict__ dcoef, float* __restrict__ csc)
{
  const int t    = threadIdx.x;
  const int lane = t & 31;
  const int wid  = t >> 5;               // 0..7 -> output N tile
  const int rowBase = blockIdx.x * 16;   // global row in [0, NB*SEQ)
  const int lo = lane & 15;
  const int hi = lane >> 4;
  const int N0 = wid * 16;

  v8f accQ = {}; v8f accK = {}; v8f accV = {};
  const float* xrow = x + (size_t)(rowBase + lo) * DIM;   // A row (M = lo)
  for (int k = 0; k < DIM; k += 4) {
    // A 16x4 f32 frag: v0 = K=0 (lanes 0-15) / K=2 (lanes 16-31); v1 = K+1
    v2f af;
    af.x = xrow[k + 2 * hi];
    af.y = xrow[k + 2 * hi + 1];
    // out = x @ W^T  =>  B[k][n] = W[n][k]
    v2f bf;
    bf.x = Qw[(size_t)(N0 + lo) * DIM + k + 2 * hi];
    bf.y = Qw[(size_t)(N0 + lo) * DIM + k + 2 * hi + 1];
    accQ = wmma4(af, bf, accQ);
    bf.x = Kw[(size_t)(N0 + lo) * DIM + k + 2 * hi];
    bf.y = Kw[(size_t)(N0 + lo) * DIM + k + 2 * hi + 1];
    accK = wmma4(af, bf, accK);
    bf.x = Vw[(size_t)(N0 + lo) * DIM + k + 2 * hi];
    bf.y = Vw[(size_t)(N0 + lo) * DIM + k + 2 * hi + 1];
    accV = wmma4(af, bf, accV);
  }
  // C/D layout: vgpr r, lane L -> element (M = r + 8*(L>>4), N = L&15)
  for (int r = 0; r < 8; ++r) {
    size_t off = (size_t)(rowBase + r + 8 * hi) * DIM + N0 + lo;
    qe[off] = __expf(accQ[r]);
    ke[off] = __expf(accK[r]);
    vv[off] = accV[r];
  }

  // Per-row scalars: a = sigmoid(x . alpha), w_s, wa, dcoef, csc.
  if (t < 16) {
    int row = rowBase + t;
    int b = row / SEQ;
    int s = row - b * SEQ;
    const float* xr = x + (size_t)row * DIM;
    float z = 0.f;
    for (int k = 0; k < DIM; ++k) z += xr[k] * alpha_w[k];
    float a_s = 1.f / (1.f + __expf(-z));
    float w0  = w0p[0];
    float w_s = 1.f / (w0 + (float)(s + 2) + 2.f);   // t = s+2, w = 1/(w0+t+2)
    wa[row] = w_s * a_s;
    int l = s + 1;                                    // padded index
    dcoef[b * LSEQ + l] = (1.f - a_s) * (1.f - w_s);
    csc[l] = w_s;                                     // batch-independent
  }
}

// ---------------------------------------------------------------------------
// K2: dense 64-step chunk product in LDS via P <- d*P + c*(Pq)q^T.
// grid = NB*SLOTS blocks; chunks >= 33 have 0 steps and emit identity.
// Dynamic LDS: 128*128 P + 128 q + 128 u  (66560 B).
// ---------------------------------------------------------------------------
__global__ __launch_bounds__(256)
void k_chunk(const float* __restrict__ qe, const float* __restrict__ dcoef,
             const float* __restrict__ csc, const float* __restrict__ w0p,
             float* __restrict__ bufA)
{
  extern __shared__ float smem[];
  float* Pm = smem;                // 128*128
  float* qv = smem + DIM * DIM;    // 128
  float* uv = qv + DIM;            // 128

  const int t = threadIdx.x;
  const int b = blockIdx.x / SLOTS;
  const int c = blockIdx.x - b * SLOTS;

  for (int m = 0; m < 64; ++m) {
    int lin = m * 256 + t;
    int i = lin >> 7, j = lin & 127;
    Pm[lin] = (i == j) ? 1.f : 0.f;
  }
  __syncthreads();

  int l0  = c * CHUNK;
  int cnt = LSEQ - l0; if (cnt < 0) cnt = 0; if (cnt > CHUNK) cnt = CHUNK;
  const float* qe_b = qe + (size_t)b * SEQ * DIM;
  const float* dc_b = dcoef + b * LSEQ;
  const float  w0   = w0p[0];

  for (int step = 0; step < cnt; ++step) {
    int l = l0 + step;
    if (t < DIM) qv[t] = (l == 0) ? 0.f : qe_b[(size_t)(l - 1) * DIM + t];
    __syncthreads();

    // u = P q : two threads per row, halves combined with a lane shuffle.
    int row = t >> 1, h = t & 1;
    const float* Pr = &Pm[row * DIM + 64 * h];
    const float* qh = &qv[64 * h];
    float partial = 0.f;
    #pragma unroll 16
    for (int j = 0; j < 64; ++j) partial += Pr[j] * qh[j];
    partial += __shfl_xor(partial, 1, 32);
    if (h == 0) uv[row] = partial;
    __syncthreads();

    float dl = (l == 0) ? 0.f : dc_b[l];   // l=0: (1-1)*(1-w0) == 0 exactly
    float cl = (l == 0) ? w0  : csc[l];
    for (int m = 0; m < 64; ++m) {
      int lin = m * 256 + t;
      int i = lin >> 7, j = lin & 127;
      Pm[lin] = dl * Pm[lin] + cl * uv[i] * qv[j];
    }
    __syncthreads();
  }

  float* outm = bufA + (size_t)(b * SLOTS + c) * (DIM * DIM);
  for (int m = 0; m < 64; ++m) {
    int lin = m * 256 + t;
    outm[lin] = Pm[lin];
  }
}

// ---------------------------------------------------------------------------
// K3: one tree level: dst[g] = src[2g] @ src[2g+1], dense 128x128x128 fp32
// WMMA GEMM. grid = NB*npairs blocks of 8 waves; wave w owns M-tile w.
// ---------------------------------------------------------------------------
__global__ __launch_bounds__(256)
void k_pairmul(const float* __restrict__ src, float* __restrict__ dst, int npairs)
{
  const int t    = threadIdx.x;
  const int lane = t & 31;
  const int wid  = t >> 5;             // M tile
  const int b = blockIdx.x / npairs;
  const int g = blockIdx.x - b * npairs;
  const float* Am = src + (size_t)(b * SLOTS + 2 * g) * (DIM * DIM);
  const float* Bm = Am + DIM * DIM;    // slot 2g+1 is contiguous
  float* Cm = dst + (size_t)(b * SLOTS + g) * (DIM * DIM);

  const int lo = lane & 15, hi = lane >> 4;
  const int M0 = wid * 16;
  v8f acc[8] = {};
  const float* Arow = Am + (size_t)(M0 + lo) * DIM;
  for (int k = 0; k < DIM; k += 4) {
    v2f af;
    af.x = Arow[k + 2 * hi];
    af.y = Arow[k + 2 * hi + 1];
    #pragma unroll
    for (int nt = 0; nt < 8; ++nt) {
      v2f bf;
      bf.x = Bm[(size_t)(k + 2 * hi) * DIM + nt * 16 + lo];
      bf.y = Bm[(size_t)(k + 2 * hi + 1) * DIM + nt * 16 + lo];
      acc[nt] = wmma4(af, bf, acc[nt]);
    }
  }
  for (int nt = 0; nt < 8; ++nt)
    for (int r = 0; r < 8; ++r)
      Cm[(size_t)(M0 + r + 8 * hi) * DIM + nt * 16 + lo] = acc[nt][r];
}

// ---------------------------------------------------------------------------
// K4: Pq = q_f @ P^T via WMMA (16-row tiles), then epilogue:
//   l==0: out = (M_0 Pq0) / (M_norm0 Pq0)  (den==0 -> 1)
//   l>=1: dk = ke . Pq; num = wa*v*dk; den = wa*dk; out = num/(den==0?1:den)
// grid = NB*129 blocks of 8 waves; wave w owns N-tile w.
// ---------------------------------------------------------------------------
__global__ __launch_bounds__(256)
void k_output(const float* __restrict__ qe, const float* __restrict__ ke,
              const float* __restrict__ vv, const float* __restrict__ wa,
              const float* __restrict__ Pfin,
              const float* __restrict__ M0m, const float* __restrict__ Mnm,
              float* __restrict__ out)
{
  __shared__ float PqT[16 * DIM];
  __shared__ float dotk[16];
  __shared__ float nd[2 * DIM];

  const int t    = threadIdx.x;
  const int lane = t & 31;
  const int wid  = t >> 5;
  const int b    = blockIdx.x / 129;
  const int tile = blockIdx.x - b * 129;
  const int lBase = tile * 16;
  const int lo = lane & 15, hi = lane >> 4;
  const int N0 = wid * 16;

  const float* qe_b = qe + (size_t)b * SEQ * DIM;
  const float* Pb   = Pfin + (size_t)b * SLOTS * DIM * DIM;   // slot 0

  int  lrow  = lBase + lo;
  bool valid = (lrow >= 1) && (lrow < LSEQ);
  const float* qrow = qe_b + (size_t)(valid ? (lrow - 1) : 0) * DIM;
  float msk = valid ? 1.f : 0.f;   // q_f[0] = 0; rows >= LSEQ masked

  v8f acc = {};
  for (int k = 0; k < DIM; k += 4) {
    v2f af;
    af.x = msk * qrow[k + 2 * hi];
    af.y = msk * qrow[k + 2 * hi + 1];
    // Pq[l][i] = sum_j q[l][j]*P[i][j] => B[k=j][n=i] = P[i*DIM + j]
    v2f bf;
    bf.x = Pb[(size_t)(N0 + lo) * DIM + k + 2 * hi];
    bf.y = Pb[(size_t)(N0 + lo) * DIM + k + 2 * hi + 1];
    acc = wmma4(af, bf, acc);
  }
  for (int r = 0; r < 8; ++r)
    PqT[(r + 8 * hi) * DIM + N0 + lo] = acc[r];
  __syncthreads();

  // dk[r] = ke[l-1] . Pq[l] : 16 lanes per row, 8 elems each, xor-reduce.
  {
    int r   = t >> 4;
    int sub = t & 15;
    int l   = lBase + r;
    float p = 0.f;
    if (l >= 1 && l < LSEQ) {
      const float* krow = ke + ((size_t)b * SEQ + (l - 1)) * DIM;
      const float* pqr  = &PqT[r * DIM];
      for (int j = sub * 8; j < sub * 8 + 8; ++j) p += krow[j] * pqr[j];
    }
    p += __shfl_xor(p, 1, 32);
    p += __shfl_xor(p, 2, 32);
    p += __shfl_xor(p, 4, 32);
    p += __shfl_xor(p, 8, 32);
    if (sub == 0) dotk[r] = p;
  }
  __syncthreads();

  for (int e = t; e < 16 * DIM; e += 256) {
    int r = e >> 7, i = e & 127;
    int l = lBase + r;
    if (l >= 1 && l < LSEQ) {
      float waL = wa[(size_t)b * SEQ + (l - 1)];
      float dk  = dotk[r];
      float num = waL * vv[((size_t)b * SEQ + (l - 1)) * DIM + i] * dk;
      float den = waL * dk;
      out[((size_t)b * LSEQ + l) * DIM + i] = num / ((den == 0.f) ? 1.f : den);
    }
  }

  if (tile == 0) {
    // row l==0: two 128x128 matvecs against Pq row 0.
    int i = t & 127, which = t >> 7;     // 0: num (M_0), 1: den (M_norm_0)
    const float* Mrow = (which ? Mnm : M0m) + (size_t)i * DIM;
    float s = 0.f;
    for (int j = 0; j < DIM; ++j) s += Mrow[j] * PqT[j];
    nd[which * DIM + i] = s;
    __syncthreads();
    if (t < DIM) {
      float den = nd[DIM + t];
      out[((size_t)b * LSEQ) * DIM + t] = nd[t] / ((den == 0.f) ? 1.f : den);
    }
  }
}

// ---------------------------------------------------------------------------
extern "C" void kernel_launch(void* const* d_in, const int* in_sizes, int n_in,
                              void* d_out, int out_size, void* d_ws, size_t ws_size,
                              hipStream_t stream)
{
  const float* x   = (const float*)d_in[0];
  const float* aw  = (const float*)d_in[1];
  const float* Qw  = (const float*)d_in[2];
  const float* Kw  = (const float*)d_in[3];
  const float* Vw  = (const float*)d_in[4];
  const float* w0  = (const float*)d_in[5];
  const float* M0m = (const float*)d_in[6];
  const float* Mnm = (const float*)d_in[7];
  float* out = (float*)d_out;

  float* ws = (float*)d_ws;
  size_t off = 0;
  float* qe    = ws + off; off += (size_t)NB * SEQ * DIM;
  float* ke    = ws + off; off += (size_t)NB * SEQ * DIM;
  float* vv    = ws + off; off += (size_t)NB * SEQ * DIM;
  float* wa    = ws + off; off += (size_t)NB * SEQ;
  float* dcoef = ws + off; off += (size_t)NB * LSEQ;
  float* csc   = ws + off; off += (size_t)LSEQ;
  off = (off + 63) & ~(size_t)63;
  float* bufA  = ws + off; off += (size_t)NB * SLOTS * DIM * DIM;
  float* bufB  = ws + off; off += (size_t)NB * SLOTS * DIM * DIM;
  // total ~5.8M floats (~23 MB) of workspace

  k_proj<<<(NB * SEQ) / 16, 256, 0, stream>>>(x, aw, Qw, Kw, Vw, w0,
                                              qe, ke, vv, wa, dcoef, csc);

  size_t lds = (size_t)(DIM * DIM + 2 * DIM) * sizeof(float);
  k_chunk<<<NB * SLOTS, 256, lds, stream>>>(qe, dcoef, csc, w0, bufA);

  float* sa = bufA;
  float* sb = bufB;
  for (int n = SLOTS / 2; n >= 1; n >>= 1) {   // 6 tree levels
    k_pairmul<<<NB * n, 256, 0, stream>>>(sa, sb, n);
    float* tp = sa; sa = sb; sb = tp;
  }
  // final P per batch at sa + b*SLOTS*DIM*DIM (slot 0)

  k_output<<<NB * 129, 256, 0, stream>>>(qe, ke, vv, wa, sa, M0m, Mnm, out);
}